// EX_Module_noselect_seq_7670811590895
// MI455X (gfx1250) — compile-verified
//
#include <hip/hip_runtime.h>
#include <hip/hip_bf16.h>

typedef float v2f __attribute__((ext_vector_type(2)));
typedef float v4f __attribute__((ext_vector_type(4)));
typedef float v8f __attribute__((ext_vector_type(8)));

#define HW_  16384
#define C_   256
#define CH_  128
#define B_   8

// GEMM tiling
#define KS_   32                 // K per pipeline stage
#define ABUF_ (KS_ * 128)        // 4096 dwords per A buffer ([kg][hw][k%4])

__device__ __forceinline__ float hsig(float v) {
    return fminf(fmaxf((v + 3.0f) * (1.0f / 6.0f), 0.0f), 1.0f);
}

// ---------------------------------------------------------------------------
// K1: per-(b,c) mean of x over hw  -> xmean (2048 floats)
// ---------------------------------------------------------------------------
__global__ __launch_bounds__(256) void k_colmean(const float* __restrict__ x,
                                                 float* __restrict__ xmean) {
    __shared__ float red[256];
    const float* p = x + (size_t)blockIdx.x * HW_;
    float s = 0.0f;
    for (int i = threadIdx.x; i < HW_; i += 256) s += p[i];
    red[threadIdx.x] = s;
    __syncthreads();
    for (int off = 128; off > 0; off >>= 1) {
        if ((int)threadIdx.x < off) red[threadIdx.x] += red[threadIdx.x + off];
        __syncthreads();
    }
    if (threadIdx.x == 0) xmean[blockIdx.x] = red[0] * (1.0f / HW_);
}

// ---------------------------------------------------------------------------
// K2: sa[b,hw] = sum_c x[b,c,hw] * w1[c] + b1
// ---------------------------------------------------------------------------
__global__ __launch_bounds__(256) void k_sa(const float* __restrict__ x,
                                            const float* __restrict__ w1,
                                            const float* __restrict__ b1,
                                            float* __restrict__ sa) {
    const int b  = blockIdx.y;
    const int hw = blockIdx.x * 256 + threadIdx.x;
    const float* p = x + (size_t)b * (C_ * HW_) + hw;
    float acc = b1[0];
#pragma unroll 8
    for (int c = 0; c < C_; ++c) acc = fmaf(p[(size_t)c * HW_], w1[c], acc);
    sa[b * HW_ + hw] = acc;
}

// ---------------------------------------------------------------------------
// K3 (tiny): g = w0 @ xmean + b0 ; ca = wb-proj ; hsigmoid ; LayerNorm(c)
// ---------------------------------------------------------------------------
__global__ __launch_bounds__(256) void k_attn(const float* __restrict__ xmean,
                                              const float* __restrict__ w0,
                                              const float* __restrict__ b0,
                                              const float* __restrict__ wb,
                                              const float* __restrict__ bb,
                                              const float* __restrict__ gamma,
                                              const float* __restrict__ beta,
                                              float* __restrict__ caOut) {
    __shared__ float G[B_ * CH_];
    __shared__ float CA[B_ * C_];
    __shared__ float MU[B_], RS[B_];
    const int tid = threadIdx.x;

    for (int i = tid; i < B_ * CH_; i += 256) {
        int b = i >> 7, o = i & (CH_ - 1);
        float s = b0[o];
        const float* wrow = w0 + o * C_;
        const float* xm   = xmean + b * C_;
        for (int c = 0; c < C_; ++c) s = fmaf(wrow[c], xm[c], s);
        G[i] = s;
    }
    __syncthreads();

    for (int i = tid; i < B_ * C_; i += 256) {
        int b = i >> 8, c = i & (C_ - 1);
        float s = bb[c];
        const float* wrow = wb + c * CH_;
        const float* gb   = G + b * CH_;
        for (int o = 0; o < CH_; ++o) s = fmaf(gb[o], wrow[o], s);
        CA[i] = hsig(s);
    }
    __syncthreads();

    if (tid < B_) {
        float mu = 0.0f;
        for (int c = 0; c < C_; ++c) mu += CA[tid * C_ + c];
        mu *= (1.0f / C_);
        float var = 0.0f;
        for (int c = 0; c < C_; ++c) {
            float d = CA[tid * C_ + c] - mu;
            var = fmaf(d, d, var);
        }
        var *= (1.0f / C_);
        MU[tid] = mu;
        RS[tid] = rsqrtf(var + 1e-5f);
    }
    __syncthreads();

    for (int i = tid; i < B_ * C_; i += 256) {
        int b = i >> 8, c = i & (C_ - 1);
        caOut[i] = (CA[i] - MU[b]) * RS[b] * gamma[c] + beta[c];
    }
}

// ---------------------------------------------------------------------------
// K4: ssum[b,c] = sum_hw hsig(ca[b,c] * sa[b,hw])
// ---------------------------------------------------------------------------
__global__ __launch_bounds__(256) void k_ssum(const float* __restrict__ sa,
                                              const float* __restrict__ ca,
                                              float* __restrict__ ssum) {
    __shared__ float red[256];
    const int bc = blockIdx.x;
    const int b  = bc >> 8;
    const float cav = ca[bc];
    const float* p = sa + b * HW_;
    float s = 0.0f;
    for (int i = threadIdx.x; i < HW_; i += 256) s += hsig(cav * p[i]);
    red[threadIdx.x] = s;
    __syncthreads();
    for (int off = 128; off > 0; off >>= 1) {
        if ((int)threadIdx.x < off) red[threadIdx.x] += red[threadIdx.x + off];
        __syncthreads();
    }
    if (threadIdx.x == 0) ssum[bc] = red[0];
}

// ---------------------------------------------------------------------------
// K5: fused main GEMM + epilogue, V_WMMA_F32_16X16X4_F32, LDS double-buffered.
//   res[b,co,hw] = sum_ci x[b,ci,hw] * wr[co,ci]
//   out = hsig(ca*sa) + ssum + res + br     (non-temporal stores)
//
// Block: 256 thr = 8 waves = 2 hw-groups x 4 N-tiles.
//   Block tile: 128 hw x 64 cout, K = 256 in 8 stages of 32.
//   Wave tile : 64 hw (4 M-tiles) x 16 cout.
// LDS (96 KB): B tile 256x64 in [k/4][n][k%4] (64 KB) +
//   double-buffered A slabs 32k x 128hw in [k/4][hw][k%4] (2 x 16 KB).
// Fragment layout => A/B operands are single aligned ds_load_b64 each,
// no register shuffles; read banks disjoint between wave halves.
// ---------------------------------------------------------------------------
__global__ __launch_bounds__(256) void k_gemm_fused(const float* __restrict__ x,
                                                    const float* __restrict__ wr,
                                                    const float* __restrict__ br,
                                                    const float* __restrict__ sa,
                                                    const float* __restrict__ ca,
                                                    const float* __restrict__ ssum,
                                                    float* __restrict__ out) {
    __shared__ float Bs[C_ * 64];        // 64 KB
    __shared__ float As[2 * ABUF_];      // 2 x 16 KB

    const int tid  = threadIdx.x;
    const int wave = tid >> 5;
    const int lane = tid & 31;
    const int nm   = lane & 15;    // N for B/C/D, M for A
    const int half = lane >> 4;

    const int wn   = wave & 3;     // N-tile within block (0..3)
    const int wm   = wave >> 2;    // hw-group within block (0..1)

    const int b         = blockIdx.z;
    const int coutGroup = blockIdx.x << 6;
    const int hwBlock   = blockIdx.y << 7;          // 128 hw per block
    const int hwBase    = hwBlock + (wm << 6);      // wave's 64-hw span

    // --- stage B tile: Bs[(k>>2)*256 + n*4 + (k&3)] = wr[coutGroup+n][k] ---
    for (int idx = tid; idx < C_ * 64; idx += 256) {
        int k = idx & (C_ - 1), nn = idx >> 8;
        Bs[((k >> 2) << 8) + (nn << 2) + (k & 3)] = wr[(coutGroup + nn) * C_ + k];
    }

    // --- cooperative A slab fill ---
    // Thread owns 4 (kGroup, hw) cells per stage; each cell: 4 coalesced b32
    // loads (k = kg*4 + 0..3 at fixed hw) packed and written as one b128 to
    // As[kg*512 + hw*4]  (lanes stride 4 dwords -> conflict-free).
    const float* xg = x + (size_t)b * (C_ * HW_) + hwBlock;
    const int kg0 = (tid + 0)   >> 7, h0 = (tid + 0)   & 127;
    const int kg1 = (tid + 256) >> 7, h1 = (tid + 256) & 127;
    const int kg2 = (tid + 512) >> 7, h2 = (tid + 512) & 127;
    const int kg3 = (tid + 768) >> 7, h3 = (tid + 768) & 127;

    v4f t0, t1, t2, t3;
#define LOAD_CELL(dst, kg, hh, ksOff)                                          \
    {                                                                          \
        const float* pc = xg + (size_t)((ksOff) + (kg) * 4) * HW_ + (hh);      \
        dst = (v4f){ pc[0], pc[HW_], pc[2 * HW_], pc[3 * HW_] };               \
    }
#define STORE_CELL(An, src, kg, hh)                                            \
    *(v4f*)&(An)[((kg) << 9) + ((hh) << 2)] = src;

    // prologue: stage 0
    LOAD_CELL(t0, kg0, h0, 0)
    LOAD_CELL(t1, kg1, h1, 0)
    LOAD_CELL(t2, kg2, h2, 0)
    LOAD_CELL(t3, kg3, h3, 0)
    STORE_CELL(As, t0, kg0, h0)
    STORE_CELL(As, t1, kg1, h1)
    STORE_CELL(As, t2, kg2, h2)
    STORE_CELL(As, t3, kg3, h3)
    __syncthreads();

    const int nloc = (wn << 4) + nm;             // local cout in [0,64)
    // A fragment base: [kg][hw][k%4] with hw = wm*64 + t*16 + nm, k%4 = half*2
    const int aoff = (((wm << 6) + nm) << 2) + (half << 1);

    v8f acc0 = {}, acc1 = {}, acc2 = {}, acc3 = {};
    int buf = 0;

#pragma unroll 1
    for (int s = 0; s < 8; ++s) {
        const int ks = s * KS_;
        if (s < 7) {  // prefetch next stage while computing this one
            LOAD_CELL(t0, kg0, h0, ks + KS_)
            LOAD_CELL(t1, kg1, h1, ks + KS_)
            LOAD_CELL(t2, kg2, h2, ks + KS_)
            LOAD_CELL(t3, kg3, h3, ks + KS_)
        }

        const float* Ab = &As[buf * ABUF_ + aoff];
#pragma unroll
        for (int k0 = 0; k0 < KS_; k0 += 4) {
            v2f bv = *(const v2f*)&Bs[((ks + k0) << 6) + (nloc << 2) + (half << 1)];
            const float* pk = Ab + ((k0 >> 2) << 9);
            v2f a0 = *(const v2f*)&pk[0];     // M-tile 0 (hw + 0)
            v2f a1 = *(const v2f*)&pk[64];    // M-tile 1 (hw + 16)
            v2f a2 = *(const v2f*)&pk[128];   // M-tile 2 (hw + 32)
            v2f a3 = *(const v2f*)&pk[192];   // M-tile 3 (hw + 48)
            acc0 = __builtin_amdgcn_wmma_f32_16x16x4_f32(false, a0, false, bv, (short)0, acc0, false, false);
            acc1 = __builtin_amdgcn_wmma_f32_16x16x4_f32(false, a1, false, bv, (short)0, acc1, false, false);
            acc2 = __builtin_amdgcn_wmma_f32_16x16x4_f32(false, a2, false, bv, (short)0, acc2, false, false);
            acc3 = __builtin_amdgcn_wmma_f32_16x16x4_f32(false, a3, false, bv, (short)0, acc3, false, false);
        }

        if (s < 7) {
            float* An = &As[(buf ^ 1) * ABUF_];
            STORE_CELL(An, t0, kg0, h0)
            STORE_CELL(An, t1, kg1, h1)
            STORE_CELL(An, t2, kg2, h2)
            STORE_CELL(An, t3, kg3, h3)
            __syncthreads();
            buf ^= 1;
        }
    }
#undef LOAD_CELL
#undef STORE_CELL

    // --- epilogue: lane nm = N; VGPR r holds M = r + half*8 ---
    const int co    = coutGroup + nloc;
    const float cav = ca[b * C_ + co];
    const float add = ssum[b * C_ + co] + br[co];
    const float* sab = sa + b * HW_;
    float* ob = out + ((size_t)(b * C_ + co)) * HW_;

    v8f accs[4] = { acc0, acc1, acc2, acc3 };
#pragma unroll
    for (int t = 0; t < 4; ++t) {
        const int hwt = hwBase + (t << 4) + (half << 3);
        v4f slo = *(const v4f*)&sab[hwt];
        v4f shi = *(const v4f*)&sab[hwt + 4];
        v8f A = accs[t];
        v4f lo, hi;
        lo.x = A[0] + add + hsig(cav * slo.x);
        lo.y = A[1] + add + hsig(cav * slo.y);
        lo.z = A[2] + add + hsig(cav * slo.z);
        lo.w = A[3] + add + hsig(cav * slo.w);
        hi.x = A[4] + add + hsig(cav * shi.x);
        hi.y = A[5] + add + hsig(cav * shi.y);
        hi.z = A[6] + add + hsig(cav * shi.z);
        hi.w = A[7] + add + hsig(cav * shi.w);
        __builtin_nontemporal_store(lo, (v4f*)&ob[hwt]);
        __builtin_nontemporal_store(hi, (v4f*)&ob[hwt + 4]);
    }
}

// ---------------------------------------------------------------------------
extern "C" void kernel_launch(void* const* d_in, const int* in_sizes, int n_in,
                              void* d_out, int out_size, void* d_ws, size_t ws_size,
                              hipStream_t stream) {
    const float* x     = (const float*)d_in[0];
    const float* w0    = (const float*)d_in[1];
    const float* b0    = (const float*)d_in[2];
    const float* wb    = (const float*)d_in[3];
    const float* bb    = (const float*)d_in[4];
    const float* w1    = (const float*)d_in[5];
    const float* b1    = (const float*)d_in[6];
    const float* wr    = (const float*)d_in[7];
    const float* br    = (const float*)d_in[8];
    const float* gamma = (const float*)d_in[9];
    const float* beta  = (const float*)d_in[10];
    float* out = (float*)d_out;

    float* ws    = (float*)d_ws;
    float* xmean = ws;            // 2048
    float* caF   = ws + 2048;     // 2048
    float* ssum  = ws + 4096;     // 2048
    float* sa    = ws + 6144;     // 131072   (total < 0.6 MB)

    k_colmean<<<B_ * C_, 256, 0, stream>>>(x, xmean);
    k_sa<<<dim3(HW_ / 256, B_), 256, 0, stream>>>(x, w1, b1, sa);
    k_attn<<<1, 256, 0, stream>>>(xmean, w0, b0, wb, bb, gamma, beta, caF);
    k_ssum<<<B_ * C_, 256, 0, stream>>>(sa, caF, ssum);
    // grid.x = cout-group (fastest) for L2 reuse of x slabs
    k_gemm_fused<<<dim3(C_ / 64, HW_ / 128, B_), 256, 0, stream>>>(x, wr, br, sa, caF, ssum, out);
}